// MetricDGNNModel_78975858639600
// MI455X (gfx1250) — compile-verified
//
#include <hip/hip_runtime.h>
#include <hip/hip_bf16.h>

typedef __attribute__((ext_vector_type(2))) float v2f;
typedef __attribute__((ext_vector_type(8))) float v8f;

#define NEG_SLOPE 0.01f

// ---------------- fill ----------------
__global__ void fill_kernel(float* __restrict__ p, float val, long n) {
    long i = (long)blockIdx.x * blockDim.x + threadIdx.x;
    if (i < n) p[i] = val;
}

// ---------------- scatter-add (segment_sum of w[e]*x[src[e]]) ----------------
// one thread per (edge, 4-float chunk); dshift = log2(feature dim)
__global__ void scatter_add_kernel(const float* __restrict__ x, const float* __restrict__ w,
                                   const int* __restrict__ src, const int* __restrict__ dst,
                                   float* __restrict__ agg, int nedges, int dshift) {
    int cshift = dshift - 2;                 // chunks per edge = d/4
    long i = (long)blockIdx.x * blockDim.x + threadIdx.x;
    long total = (long)nedges << cshift;
    if (i >= total) return;
    int e = (int)(i >> cshift);
    int c = (int)(i & ((1 << cshift) - 1));
    int s = src[e], d = dst[e];
    float wv = w[e];
    const float4* xp = (const float4*)(x + ((size_t)s << dshift)) + c;
    float4 v = *xp;
    float* ap = agg + ((size_t)d << dshift) + ((size_t)c << 2);
    atomicAdd(ap + 0, v.x * wv);
    atomicAdd(ap + 1, v.y * wv);
    atomicAdd(ap + 2, v.z * wv);
    atomicAdd(ap + 3, v.w * wv);
}

// ---------------- scatter-max ----------------
__device__ __forceinline__ void atomicMaxF(float* addr, float val) {
    // classic monotone bit trick: works vs. -inf init, mixed signs, no NaNs here
    if (!(val < 0.0f)) {
        atomicMax((int*)addr, __float_as_int(val));
    } else {
        atomicMin((unsigned int*)addr, __float_as_uint(val));
    }
}

__global__ void scatter_max_kernel(const float* __restrict__ x, const float* __restrict__ w,
                                   const int* __restrict__ src, const int* __restrict__ dst,
                                   float* __restrict__ agg, int nedges, int dshift) {
    int cshift = dshift - 2;
    long i = (long)blockIdx.x * blockDim.x + threadIdx.x;
    long total = (long)nedges << cshift;
    if (i >= total) return;
    int e = (int)(i >> cshift);
    int c = (int)(i & ((1 << cshift) - 1));
    int s = src[e], d = dst[e];
    float wv = w[e];
    const float4* xp = (const float4*)(x + ((size_t)s << dshift)) + c;
    float4 v = *xp;
    float* ap = agg + ((size_t)d << dshift) + ((size_t)c << 2);
    atomicMaxF(ap + 0, v.x * wv);
    atomicMaxF(ap + 1, v.y * wv);
    atomicMaxF(ap + 2, v.z * wv);
    atomicMaxF(ap + 3, v.w * wv);
}

// ---------------- fused dual-GEMM: out = leaky(A@Wr + X@Wo + b) ----------------
// fp32 WMMA 16x16x4; one wave computes one 16x16 tile. M%16==0, N%16==0, K%4==0.
// A-frag layout (16x4 f32): lanes 0-15 hold (K=k,k+1), lanes 16-31 hold (K=k+2,k+3), row = lane&15.
// B-frag layout (4x16 f32): lanes 0-15 col=lane, v0=W[k][n], v1=W[k+1][n]; lanes 16-31: K+2,K+3.
// C/D: VGPR r -> row = r + 8*(lane>=16), col = lane&15.
__global__ void gemm_dual_wmma_kernel(const float* __restrict__ A, const float* __restrict__ X,
                                      const float* __restrict__ Wr, const float* __restrict__ Wo,
                                      const float* __restrict__ bias, float* __restrict__ out,
                                      int M, int K, int N, int sanitizeA) {
    int wave = (int)(((long)blockIdx.x * blockDim.x + threadIdx.x) >> 5);
    int lane = threadIdx.x & 31;
    int tilesN = N >> 4;
    int totalTiles = (M >> 4) * tilesN;
    if (wave >= totalTiles) return;          // wave-uniform exit: EXEC stays all-ones below

    int tmi = wave / tilesN;
    int tm = tmi << 4;
    int tn = (wave - tmi * tilesN) << 4;
    int hi = lane >> 4;                      // 0: K pair {0,1}, 1: K pair {2,3}
    int lr = lane & 15;

    const float* Ap  = A  + (size_t)(tm + lr) * K + (hi << 1);
    const float* Xp  = X  + (size_t)(tm + lr) * K + (hi << 1);
    const float* Wrp = Wr + (size_t)(hi << 1) * N + tn + lr;
    const float* Wop = Wo + (size_t)(hi << 1) * N + tn + lr;

    v8f c = {};
    for (int k = 0; k < K; k += 4) {
        float a0 = Ap[k], a1 = Ap[k + 1];
        if (sanitizeA) {                     // segment_max empty segments: -inf -> 0
            a0 = __builtin_isinf(a0) ? 0.0f : a0;
            a1 = __builtin_isinf(a1) ? 0.0f : a1;
        }
        v2f av = {a0, a1};
        v2f xv = {Xp[k], Xp[k + 1]};
        size_t wofs = (size_t)k * N;
        v2f bw = {Wrp[wofs], Wrp[wofs + N]};
        v2f bo = {Wop[wofs], Wop[wofs + N]};
        c = __builtin_amdgcn_wmma_f32_16x16x4_f32(false, av, false, bw, (short)0, c, false, false);
        c = __builtin_amdgcn_wmma_f32_16x16x4_f32(false, xv, false, bo, (short)0, c, false, false);
    }

    float bcol = bias[tn + lr];
    float* op = out + (size_t)(tm + (hi << 3)) * N + tn + lr;
#pragma unroll
    for (int r = 0; r < 8; ++r) {
        float v = c[r] + bcol;
        v = (v > 0.0f) ? v : NEG_SLOPE * v;
        op[(size_t)r * N] = v;
    }
}

// ---------------- driver ----------------
static inline int cdiv_l(long a, int b) { return (int)((a + b - 1) / b); }

extern "C" void kernel_launch(void* const* d_in, const int* in_sizes, int n_in,
                              void* d_out, int out_size, void* d_ws, size_t ws_size,
                              hipStream_t stream) {
    const float* x_metric = (const float*)d_in[0];
    const float* x_alert  = (const float*)d_in[1];
    const float* w_corr   = (const float*)d_in[2];
    const float* w_cause  = (const float*)d_in[3];
    const int* src_corr   = (const int*)d_in[4];
    const int* dst_corr   = (const int*)d_in[5];
    const int* src_cause  = (const int*)d_in[6];
    const int* dst_cause  = (const int*)d_in[7];
    const float* Wr_c0 = (const float*)d_in[8];
    const float* br_c0 = (const float*)d_in[9];
    const float* Wo_c0 = (const float*)d_in[10];
    const float* Wr_k0 = (const float*)d_in[11];
    const float* br_k0 = (const float*)d_in[12];
    const float* Wo_k0 = (const float*)d_in[13];
    // d_in[14..16] = Wr_c1/br_c1/Wo_c1: dead (layer-1 metric update is never read)
    const float* Wr_k1 = (const float*)d_in[17];
    const float* br_k1 = (const float*)d_in[18];
    const float* Wo_k1 = (const float*)d_in[19];

    const int F   = 64;
    const int N_M = in_sizes[0] / F;
    const int N_A = in_sizes[1] / F;
    const int E_C = in_sizes[2];
    const int E_K = in_sizes[3];
    const int D0 = 128, D1 = 256;

    // workspace layout (floats)
    float* ws     = (float*)d_ws;
    float* agg_c0 = ws;                              // N_M * 64
    float* xm1    = agg_c0 + (size_t)N_M * F;        // N_M * 128
    float* agg_k0 = xm1    + (size_t)N_M * D0;       // N_A * 64
    float* xa1    = agg_k0 + (size_t)N_A * F;        // N_A * 128
    float* agg_k1 = xa1    + (size_t)N_A * D0;       // N_A * 128

    const int TB = 256;
    const float NEG_INF = -__builtin_huge_valf();

    // init aggregation buffers (every call: deterministic)
    {
        long n = (long)N_M * F;
        fill_kernel<<<cdiv_l(n, TB), TB, 0, stream>>>(agg_c0, 0.0f, n);
        n = (long)N_A * F;
        fill_kernel<<<cdiv_l(n, TB), TB, 0, stream>>>(agg_k0, NEG_INF, n);
        n = (long)N_A * D0;
        fill_kernel<<<cdiv_l(n, TB), TB, 0, stream>>>(agg_k1, NEG_INF, n);
    }

    // ---- layer 0: metric <- metric (sum), d=64 (shift 6, 16 chunks/edge)
    scatter_add_kernel<<<cdiv_l((long)E_C * 16, TB), TB, 0, stream>>>(
        x_metric, w_corr, src_corr, dst_corr, agg_c0, E_C, 6);

    // ---- layer 0: alert <- metric (max), d=64
    scatter_max_kernel<<<cdiv_l((long)E_K * 16, TB), TB, 0, stream>>>(
        x_metric, w_cause, src_cause, dst_cause, agg_k0, E_K, 6);

    // ---- layer 0 GEMMs: xm1 = leaky(agg_c0@Wr_c0 + x_metric@Wo_c0 + br_c0)  [N_M x 128]
    {
        long waves = (long)(N_M >> 4) * (D0 >> 4);
        gemm_dual_wmma_kernel<<<cdiv_l(waves * 32, TB), TB, 0, stream>>>(
            agg_c0, x_metric, Wr_c0, Wo_c0, br_c0, xm1, N_M, F, D0, 0);
    }
    //                  xa1 = leaky(agg_k0@Wr_k0 + x_alert@Wo_k0 + br_k0)      [N_A x 128]
    {
        long waves = (long)(N_A >> 4) * (D0 >> 4);
        gemm_dual_wmma_kernel<<<cdiv_l(waves * 32, TB), TB, 0, stream>>>(
            agg_k0, x_alert, Wr_k0, Wo_k0, br_k0, xa1, N_A, F, D0, 1);
    }

    // ---- layer 1: alert <- metric (max), d=128 (shift 7, 32 chunks/edge)
    scatter_max_kernel<<<cdiv_l((long)E_K * 32, TB), TB, 0, stream>>>(
        xm1, w_cause, src_cause, dst_cause, agg_k1, E_K, 7);

    // ---- layer 1 GEMM: out = leaky(agg_k1@Wr_k1 + xa1@Wo_k1 + br_k1)        [N_A x 256]
    {
        long waves = (long)(N_A >> 4) * (D1 >> 4);
        gemm_dual_wmma_kernel<<<cdiv_l(waves * 32, TB), TB, 0, stream>>>(
            agg_k1, xa1, Wr_k1, Wo_k1, br_k1, (float*)d_out, N_A, D0, D1, 1);
    }
}